// ProtoRNN_3899830305315
// MI455X (gfx1250) — compile-verified
//
#include <hip/hip_runtime.h>
#include <math.h>

// ---------------- problem constants (from reference) ----------------
#define N_EL        4096
#define T_STEPS     40
#define TAU_STIM_C  20
#define WRP_MAX_C   0.0625f            // 25/400
#define WC_C        5.0f               // WEE * N_NEIGH
#define INV_TAU_UE  (1.0f/20.0f)
#define INV_TAU_UI  (1.0f/10.0f)
#define INV_TAU_TH  1.0e-7f
#define INV_TAU_WRP (1.0f/2.0e9f)
#define DECAY_C     0.31622776601683794f   // 0.1^(1/2)
#define MWORDS      (N_EL / 32)            // mask bit-words per row

typedef float v2f __attribute__((ext_vector_type(2)));
typedef float v8f __attribute__((ext_vector_type(8)));

// -------- one-time mask bit-packing: 67 MB f32 -> 2 MB bits ---------
// wave32 ballot: each lane tests one mask element; lane0 stores the word.
__global__ void pack_mask_kernel(const float* __restrict__ mask,
                                 unsigned* __restrict__ mbits) {
    size_t idx = (size_t)blockIdx.x * blockDim.x + threadIdx.x;
    float v = mask[idx];
    unsigned w = __builtin_amdgcn_ballot_w32(v != 0.0f);
    if ((threadIdx.x & 31) == 0) mbits[idx >> 5] = w;
}

// -------- per-step rates: re=relu(ue)^2, ri=relu(ui)^2, hebb row coef --------
__global__ void rates_kernel(const float* __restrict__ theta,
                             float* __restrict__ ue, float* __restrict__ ui,
                             float* __restrict__ re, float* __restrict__ ri,
                             float* __restrict__ acoef,
                             int reset, float hebb_scale) {
    int i = blockIdx.x * blockDim.x + threadIdx.x;
    float u = reset ? 0.0f : ue[i];
    float v = reset ? 0.0f : ui[i];
    if (reset) { ue[i] = 0.0f; ui[i] = 0.0f; }
    float r = fmaxf(u, 0.0f); r = r * r;     // BETA = 1
    float q = fmaxf(v, 0.0f); q = q * q;
    re[i] = r; ri[i] = q;
    float th = theta[i];
    // a_i = re_i*(re_i-theta_i)/theta_i * dec * DT / TAU_WRP
    acoef[i] = r * (r - th) / th * hebb_scale;
}

// -------- stimulus-onset row renormalization: wrp_row *= WC / sum(row) -------
__global__ void renorm_kernel(float* __restrict__ wrp) {
    __shared__ float red[256];
    const int row = blockIdx.x;
    float* p = wrp + (size_t)row * N_EL;
    float s = 0.0f;
    for (int j = threadIdx.x; j < N_EL; j += 256) s += p[j];
    red[threadIdx.x] = s;
    __syncthreads();
    for (int st = 128; st > 0; st >>= 1) {
        if (threadIdx.x < st) red[threadIdx.x] += red[threadIdx.x + st];
        __syncthreads();
    }
    float scale = WC_C / red[0];
    for (int j = threadIdx.x; j < N_EL; j += 256) p[j] *= scale;
}

// ---------------- fused streaming step kernel ----------------
// One block owns 16 rows. 4 waves split the 4096-wide K dimension (1024 each).
// Outer loop walks 32-column mask words; inner 8x-unrolled loop does one
// V_WMMA_F32_16X16X4_F32 per 4 columns:
//   * wrp tile -> WMMA A, re broadcast across B's 16 columns (all accumulator
//     columns hold the same row-dot partial; take column 0 at the end),
//   * masked Hebbian clip-update written back to wrp (matvec uses pre-update w),
//   * wie0.re and wei0.ri partials on VALU.
// A-fragment (ISA 7.12.2, 32-bit A 16x4): lane m -> row m cols k,k+1;
// lane m+16 -> row m cols k+2,k+3. B 4x16: VGPR0 = {K0 | K2}, VGPR1 = {K1 | K3}.
__global__ __launch_bounds__(128)
void step_main_kernel(float* __restrict__ wrp,
                      const unsigned* __restrict__ mbits,
                      const float* __restrict__ wei0,
                      const float* __restrict__ wie0,
                      const float* __restrict__ re,
                      const float* __restrict__ ri,
                      const float* __restrict__ acoef,
                      const float* __restrict__ rt,
                      float* __restrict__ ue, float* __restrict__ ui,
                      float* __restrict__ theta,
                      float* __restrict__ ys_t) {
    __shared__ float swrp[16], swie[16], swei[16];
    const int tid  = threadIdx.x;
    const int lane = tid & 31;
    const int wv   = tid >> 5;            // 4 waves
    const int m    = lane & 15;           // row within 16-row tile
    const int koff = (lane >> 4) << 1;    // 0 or 2 (A-fragment half-wave split)
    const int row0 = blockIdx.x * 16;
    const int row  = row0 + m;
    if (tid < 16) { swrp[tid] = 0.0f; swie[tid] = 0.0f; swei[tid] = 0.0f; }
    __syncthreads();

    const float  am    = acoef[row];
    const size_t rbase = (size_t)row * N_EL;
    v8f   c = {0.f,0.f,0.f,0.f,0.f,0.f,0.f,0.f};
    float accwie = 0.0f, accwei = 0.0f;

    const int k_begin = wv * (N_EL / 4);
    const int k_end   = k_begin + (N_EL / 4);
    for (int kw = k_begin; kw < k_end; kw += 32) {
        const unsigned mw = mbits[(size_t)row * MWORDS + (kw >> 5)];
        #pragma unroll
        for (int j = 0; j < 8; ++j) {
            const int bsh = 4 * j + koff;
            const int col = kw + bsh;
            v2f w   = *(const v2f*)(wrp  + rbase + col);
            v2f b   = *(const v2f*)(re   + col);      // B fragment (broadcast)
            v2f rib = *(const v2f*)(ri   + col);
            v2f we  = *(const v2f*)(wei0 + rbase + col);
            v2f wi  = *(const v2f*)(wie0 + rbase + col);
            // row-dot partials of wrp@re on the matrix core (f32 WMMA)
            c = __builtin_amdgcn_wmma_f32_16x16x4_f32(false, w, false, b,
                                                      (short)0, c, false, false);
            accwie += wi.x * b.x   + wi.y * b.y;
            accwei += we.x * rib.x + we.y * rib.y;
            // masked Hebbian update with clip [0, WRP_MAX]
            const float mk0 = ((mw >> bsh) & 1u)       ? 1.0f : 0.0f;
            const float mk1 = ((mw >> (bsh + 1)) & 1u) ? 1.0f : 0.0f;
            v2f wn;
            wn.x = fminf(fmaxf(w.x + am * b.x * mk0, 0.0f), WRP_MAX_C);
            wn.y = fminf(fmaxf(w.y + am * b.y * mk1, 0.0f), WRP_MAX_C);
            *(v2f*)(wrp + rbase + col) = wn;
        }
    }

    // cross-wave reduction of the three row-dots
    atomicAdd(&swie[m], accwie);
    atomicAdd(&swei[m], accwei);
    if ((lane & 15) == 0) {               // lane 0 -> rows 0..7, lane 16 -> 8..15
        const int base = (lane >> 4) * 8;
        #pragma unroll
        for (int v = 0; v < 8; ++v) atomicAdd(&swrp[base + v], c[v]);
    }
    __syncthreads();

    // per-row ODE epilogue: this block exclusively owns rows row0..row0+15
    if (tid < 16) {
        const int r_ = row0 + tid;
        float swrpv = swrp[tid], swiev = swie[tid], sweiv = swei[tid];
        float u  = ue[r_];
        float un = fmaxf(u + (-u + swrpv - sweiv + rt[r_]) * INV_TAU_UE, 0.0f);
        float v  = ui[r_];
        float vn = fmaxf(v + (-v + swiev) * INV_TAU_UI, 0.0f);
        ue[r_] = un; ui[r_] = vn;
        float rr = re[r_], th = theta[r_];
        theta[r_] = th + (-th + rr * rr) * INV_TAU_TH;
        ys_t[r_] = un;
        ys_t[N_EL + r_] = vn;
    }
}

extern "C" void kernel_launch(void* const* d_in, const int* in_sizes, int n_in,
                              void* d_out, int out_size, void* d_ws, size_t ws_size,
                              hipStream_t stream) {
    (void)in_sizes; (void)n_in; (void)out_size; (void)ws_size;
    const float* r_in   = (const float*)d_in[0];   // [2,1,4096]
    const float* theta0 = (const float*)d_in[1];   // [4096]
    const float* wrp0   = (const float*)d_in[2];   // [4096,4096]
    const float* mask   = (const float*)d_in[3];   // [4096,4096]
    const float* wei0   = (const float*)d_in[4];   // [4096,4096]
    const float* wie0   = (const float*)d_in[5];   // [4096,4096]
    float* ys = (float*)d_out;                     // [40, 8192]

    // workspace layout
    char*     ws    = (char*)d_ws;
    float*    wrp   = (float*)ws;  ws += (size_t)N_EL * N_EL * sizeof(float);
    unsigned* mbits = (unsigned*)ws; ws += (size_t)N_EL * MWORDS * sizeof(unsigned);
    float*    theta = (float*)ws;  ws += N_EL * sizeof(float);
    float*    ue    = (float*)ws;  ws += N_EL * sizeof(float);
    float*    ui    = (float*)ws;  ws += N_EL * sizeof(float);
    float*    re    = (float*)ws;  ws += N_EL * sizeof(float);
    float*    ri    = (float*)ws;  ws += N_EL * sizeof(float);
    float*    acoef = (float*)ws;  ws += N_EL * sizeof(float);

    // wrp / theta are evolving state: copy inputs into workspace (D2D, async)
    hipMemcpyAsync(wrp,   wrp0,   (size_t)N_EL * N_EL * sizeof(float),
                   hipMemcpyDeviceToDevice, stream);
    hipMemcpyAsync(theta, theta0, N_EL * sizeof(float),
                   hipMemcpyDeviceToDevice, stream);
    // one-time mask bit-packing (saves ~67 MB of HBM traffic per step)
    pack_mask_kernel<<<(N_EL * (size_t)N_EL) / 256, 256, 0, stream>>>(mask, mbits);

    for (int t = 0; t < T_STEPS; ++t) {
        const int   reset      = (t % TAU_STIM_C) == 0;
        const float dec        = (t / TAU_STIM_C) == 0 ? 1.0f : DECAY_C;
        const float hebb_scale = dec * INV_TAU_WRP;        // * DT (=1)
        if (reset)
            renorm_kernel<<<N_EL, 256, 0, stream>>>(wrp);
        rates_kernel<<<N_EL / 256, 256, 0, stream>>>(theta, ue, ui, re, ri,
                                                     acoef, reset, hebb_scale);
        const float* rt = r_in + (size_t)(t / TAU_STIM_C) * N_EL;
        step_main_kernel<<<N_EL / 16, 128, 0, stream>>>(
            wrp, mbits, wei0, wie0, re, ri, acoef, rt, ue, ui, theta,
            ys + (size_t)t * 2 * N_EL);
    }
}